// DataEmbedder_39874476376348
// MI455X (gfx1250) — compile-verified
//
#include <hip/hip_runtime.h>

// DataEmbedder: 4 categorical gathers (32/64/16/8) + 8 numeric passthrough -> [B*T, 128] f32.
// Memory-bound: ~147 MB compulsory HBM traffic -> ~6.3 us floor @ 23.3 TB/s.
// CDNA5 paths used: global_load_async_to_lds_b128 (+ s_wait_asynccnt) to stage the two
// small embedding tables into LDS, ds_load_b128 (via asm, so the optimizer cannot fold
// LDS reads it cannot prove are preceded by visible stores), global_prefetch for the
// streamed dataset rows, non-temporal b128 stores for the 134 MB output stream.

typedef __attribute__((ext_vector_type(4))) float v4f;  // native vector: OK for nontemporal builtins

#define NROWS (64 * 4096)          // B*T
#define EMB2_F4 800                // 200*16 floats / 4
#define EMB3_F4 100                // 50*8 floats / 4
#define STAGE_F4 (EMB2_F4 + EMB3_F4)
#define SMEM_FLOATS 3600           // 14.4 KB LDS
#define EMB3_SMEM_BASE 3200        // float offset of emb3 inside smem

// Volatile LDS 128-bit read: ds_load_b128 from a 32-bit LDS byte address, then wait
// on DScnt before the value is used. Volatile asm with a register result cannot be
// eliminated by the optimizer (the LDS data was written by the async copy engine,
// which the compiler cannot see).
__device__ __forceinline__ v4f lds_read_b128(unsigned lds_byte_addr) {
    v4f r;
    asm volatile("ds_load_b128 %0, %1\n\t"
                 "s_wait_dscnt 0x0"
                 : "=v"(r) : "v"(lds_byte_addr) : "memory");
    return r;
}

__global__ __launch_bounds__(256) void DataEmbedder_kernel(
    const float* __restrict__ dataset,
    const float* __restrict__ emb0, const int* __restrict__ lut0,
    const float* __restrict__ emb1, const int* __restrict__ lut1,
    const float* __restrict__ emb2, const int* __restrict__ lut2,
    const float* __restrict__ emb3, const int* __restrict__ lut3,
    float* __restrict__ out)
{
    __shared__ float smem[SMEM_FLOATS];

    // ---- Stage emb2 (12.8 KB) + emb3 (1.6 KB) into LDS via async copy engine ----
    // Low 32 bits of the flat address of a __shared__ object are the LDS byte offset.
    unsigned smem_base = (unsigned)(uintptr_t)(&smem[0]);
    for (unsigned i = threadIdx.x; i < STAGE_F4; i += blockDim.x) {
        unsigned lds_off = smem_base + i * 16u;
        unsigned long long src = (i < EMB2_F4)
            ? (unsigned long long)(uintptr_t)(emb2 + (size_t)i * 4)
            : (unsigned long long)(uintptr_t)(emb3 + (size_t)(i - EMB2_F4) * 4);
        asm volatile("global_load_async_to_lds_b128 %0, %1, off"
                     :: "v"(lds_off), "v"(src)
                     : "memory");
    }
    asm volatile("s_wait_asynccnt 0x0" ::: "memory");
    __syncthreads();

    // ---- One wave per output row: 32 lanes x float4 = 128 floats = 512 B store ----
    const unsigned lane = threadIdx.x & 31u;
    const unsigned waveInBlk = threadIdx.x >> 5;
    const unsigned wavesPerBlk = blockDim.x >> 5;
    const unsigned gw = blockIdx.x * wavesPerBlk + waveInBlk;
    const unsigned nw = gridDim.x * wavesPerBlk;

    for (unsigned row = gw; row < NROWS; row += nw) {
        const float* drow = dataset + (size_t)row * 12;

        // prefetch the dataset row this wave touches next (speculative, dropped if OOB)
        __builtin_prefetch(dataset + ((size_t)row + nw) * 12, 0, 0);

        v4f v;
        if (lane < 8u) {                      // out cols [0,32): emb0, dim 32
            int id = lut0[(int)drow[0]];
            v = *(const v4f*)(emb0 + (size_t)id * 32 + lane * 4);
        } else if (lane < 24u) {              // out cols [32,96): emb1, dim 64
            int id = lut1[(int)drow[1]];
            v = *(const v4f*)(emb1 + (size_t)id * 64 + (lane - 8u) * 4);
        } else if (lane < 28u) {              // out cols [96,112): emb2, dim 16 (LDS)
            int id = lut2[(int)drow[2]];
            v = lds_read_b128(smem_base + (unsigned)(id * 16 + (lane - 24u) * 4) * 4u);
        } else if (lane < 30u) {              // out cols [112,120): emb3, dim 8 (LDS)
            int id = lut3[(int)drow[3]];
            v = lds_read_b128(smem_base + (unsigned)(EMB3_SMEM_BASE + id * 8 + (lane - 28u) * 4) * 4u);
        } else {                              // out cols [120,128): numeric cols 4..11
            v = *(const v4f*)(drow + 4 + (lane - 30u) * 4);  // 16B aligned: 48*row+16/+32
        }

        // streaming store: keep embedding tables resident in L2
        __builtin_nontemporal_store(v, (v4f*)(out + (size_t)row * 128 + lane * 4));
    }
}

extern "C" void kernel_launch(void* const* d_in, const int* in_sizes, int n_in,
                              void* d_out, int out_size, void* d_ws, size_t ws_size,
                              hipStream_t stream) {
    // setup_inputs() dict order: dataset, emb0, lut0, emb1, lut1, emb2, lut2, emb3, lut3
    const float* dataset = (const float*)d_in[0];
    const float* emb0    = (const float*)d_in[1];
    const int*   lut0    = (const int*)  d_in[2];
    const float* emb1    = (const float*)d_in[3];
    const int*   lut1    = (const int*)  d_in[4];
    const float* emb2    = (const float*)d_in[5];
    const int*   lut2    = (const int*)  d_in[6];
    const float* emb3    = (const float*)d_in[7];
    const int*   lut3    = (const int*)  d_in[8];
    float* out = (float*)d_out;

    // 2048 blocks x 8 waves = 16384 waves, 16 rows each; amortizes LDS staging,
    // saturates HBM write bandwidth.
    dim3 grid(2048), block(256);
    hipLaunchKernelGGL(DataEmbedder_kernel, grid, block, 0, stream,
                       dataset, emb0, lut0, emb1, lut1, emb2, lut2, emb3, lut3, out);
}